// DGLMultiHeadHeteroGraphLayer_33346126086309
// MI455X (gfx1250) — compile-verified
//
#include <hip/hip_runtime.h>
#include <hip/hip_bf16.h>

// ---------------------------------------------------------------------------
// DGL 2-layer hetero GAT for MI455X (gfx1250).
// GEMMs use V_WMMA_F32_16X16X4_F32 (exact fp32 WMMA). Edge softmax/scatter is
// the bandwidth-bound part and uses flipped-uint atomicMax + f32 atomicAdd.
// ---------------------------------------------------------------------------

typedef __attribute__((ext_vector_type(2))) float v2f;
typedef __attribute__((ext_vector_type(8))) float v8f;

#define HH   4      // heads
#define DDIM 64     // per-head dim
#define CDIM 256    // HH*DDIM == IN_DIM == hidden width (same every layer)

// ---------------- small helpers ----------------
__device__ __forceinline__ float leaky02(float x) { return x > 0.f ? x : 0.2f * x; }

// monotone float <-> uint mapping so unsigned atomicMax == float max
__device__ __forceinline__ unsigned flipf(float f) {
  unsigned u = __float_as_uint(f);
  return (u & 0x80000000u) ? ~u : (u | 0x80000000u);
}
__device__ __forceinline__ float unflipf(unsigned u) {
  u = (u & 0x80000000u) ? (u & 0x7fffffffu) : ~u;
  return __uint_as_float(u);
}

// ---------------- utility kernels ----------------
__global__ void k_zero(float* __restrict__ p, int n) {
  int i = blockIdx.x * blockDim.x + threadIdx.x;
  if (i < n) p[i] = 0.f;
}

__global__ void k_init_emax(unsigned* __restrict__ p, int n) {
  int i = blockIdx.x * blockDim.x + threadIdx.x;
  if (i < n) p[i] = 0x007FFFFFu;   // == flipf(-inf)
}

// ---------------- WMMA fp32 GEMM:  C[M,256] = A[M,256] @ B[256,256] ----------
// grid.x = M/16, grid.y = 2 ; 256 threads = 8 waves, each wave one 16x16 tile.
__global__ void __launch_bounds__(256)
k_gemm_wmma_f32(const float* __restrict__ A, const float* __restrict__ B,
                float* __restrict__ C, int M) {
  const int K = CDIM, N = CDIM;
  const int wave = threadIdx.x >> 5;
  const int lane = threadIdx.x & 31;
  const int lo = lane & 15;          // row within A tile / col within B,C tile
  const int hi = lane >> 4;          // half-wave select
  const int row0 = blockIdx.x << 4;
  const int col0 = ((blockIdx.y << 3) + wave) << 4;
  if (row0 >= M) return;             // uniform per block: EXEC stays all-ones

  v8f c = {};
  // A 16x4 tile layout: lanes 0-15 hold M=lo, vgpr0/1 = K {0,1}; lanes 16-31 K {2,3}
  const float* arow = A + (size_t)(row0 + lo) * K + 2 * hi;
  // B 4x16 tile layout mirrored on N: vgpr0/1 = K {0,1} (lanes 0-15) / {2,3}
  const float* bcol = B + (size_t)(2 * hi) * N + col0 + lo;

  for (int k = 0; k < K; k += 4) {
    v2f a = *(const v2f*)(arow + k);       // K = k+2hi, k+2hi+1 (8B aligned)
    v2f b;
    b.x = bcol[(size_t)k * N];             // K = k+2hi
    b.y = bcol[(size_t)k * N + N];         // K = k+2hi+1
    // 8-arg form: (neg_a, A, neg_b, B, c_mod, C, reuse_a, reuse_b)
    c = __builtin_amdgcn_wmma_f32_16x16x4_f32(false, a, false, b,
                                              (short)0, c, false, false);
  }

  // C layout: vgpr i -> M = row0 + i + 8*hi, N = col0 + lo
  float* crow = C + (size_t)(row0 + 8 * hi) * N + col0 + lo;
#pragma unroll
  for (int i = 0; i < 8; ++i) crow[(size_t)i * N] = c[i];
}

// ---------------- war[k,h] = sum_d W[k, h*D+d] * ar[h,d] ----------------
__global__ void k_war(const float* __restrict__ W, const float* __restrict__ ar,
                      float* __restrict__ war) {
  int t = blockIdx.x * blockDim.x + threadIdx.x;   // t = k*HH + h
  if (t >= CDIM * HH) return;
  int k = t >> 2, h = t & 3;
  const float* wrow = W + (size_t)k * CDIM + h * DDIM;
  const float* arh  = ar + h * DDIM;
  float s = 0.f;
#pragma unroll 8
  for (int d = 0; d < DDIM; ++d) s += wrow[d] * arh[d];
  war[t] = s;
}

// ---------------- el[n,h] = sum_d hs[n,h,d] * al[h,d] ----------------
__global__ void k_el(const float* __restrict__ hs, const float* __restrict__ al,
                     float* __restrict__ el, int N) {
  int t = blockIdx.x * blockDim.x + threadIdx.x;   // t = n*HH + h
  if (t >= N * HH) return;
  int n = t >> 2, h = t & 3;
  const float* hrow = hs + (size_t)n * CDIM + h * DDIM;
  const float* alh  = al + h * DDIM;
  float s = 0.f;
#pragma unroll 8
  for (int d = 0; d < DDIM; ++d) s += hrow[d] * alh[d];
  el[t] = s;
}

// ---------------- er[n,h] = sum_k feat[n,k] * war[k,h] ----------------
__global__ void k_er(const float* __restrict__ feat, const float* __restrict__ war,
                     float* __restrict__ er, int N) {
  int t = blockIdx.x * blockDim.x + threadIdx.x;   // t = n*HH + h
  if (t >= N * HH) return;
  int n = t >> 2, h = t & 3;
  const float* f = feat + (size_t)n * CDIM;
  float s = 0.f;
#pragma unroll 4
  for (int k = 0; k < CDIM; ++k) s += f[k] * war[k * HH + h];
  er[t] = s;
}

// ---------------- edge pass 1: per-dst segment max ----------------
__global__ void k_edge_max(const int* __restrict__ src, const int* __restrict__ dst,
                           const float* __restrict__ el, const float* __restrict__ er,
                           unsigned* __restrict__ emax, int E) {
  int i = blockIdx.x * blockDim.x + threadIdx.x;
  if (i >= E) return;
  int s = src[i], d = dst[i];
#pragma unroll
  for (int h = 0; h < HH; ++h) {
    float e = leaky02(el[s * HH + h] + er[d * HH + h]);
    atomicMax(&emax[d * HH + h], flipf(e));
  }
}

// ---------------- edge pass 2: softmax denominator ----------------
__global__ void k_edge_denom(const int* __restrict__ src, const int* __restrict__ dst,
                             const float* __restrict__ el, const float* __restrict__ er,
                             const unsigned* __restrict__ emax,
                             float* __restrict__ denom, int E) {
  int i = blockIdx.x * blockDim.x + threadIdx.x;
  if (i >= E) return;
  int s = src[i], d = dst[i];
#pragma unroll
  for (int h = 0; h < HH; ++h) {
    float e = leaky02(el[s * HH + h] + er[d * HH + h]);
    float m = unflipf(emax[d * HH + h]);
    atomicAdd(&denom[d * HH + h], expf(e - m));
  }
}

// ---------------- edge pass 3: weighted scatter-add ----------------
// one 256-thread block per edge: thread t -> (h = t>>6, d = t&63)
__global__ void __launch_bounds__(256)
k_edge_scatter(const int* __restrict__ src, const int* __restrict__ dst,
               const float* __restrict__ el, const float* __restrict__ er,
               const unsigned* __restrict__ emax, const float* __restrict__ denom,
               const float* __restrict__ hs, float* __restrict__ accum, int E) {
  int edge = blockIdx.x;
  if (edge >= E) return;
  int t = threadIdx.x;
  int h = t >> 6, d = t & 63;
  int s = src[edge], dn = dst[edge];
  float e   = leaky02(el[s * HH + h] + er[dn * HH + h]);   // broadcast loads (L0 hit)
  float m   = unflipf(emax[dn * HH + h]);
  float den = denom[dn * HH + h];
  float alpha = expf(e - m) / fmaxf(den, 1e-9f);
  float v = hs[(size_t)s * CDIM + h * DDIM + d] * alpha;
  atomicAdd(&accum[(size_t)dn * CDIM + h * DDIM + d], v);
}

// ---------------- bias (sum of the two contributing etype biases) + ReLU ----
__global__ void k_finalize(float* __restrict__ x, const float* __restrict__ bA,
                           const float* __restrict__ bB, int N) {
  int i = blockIdx.x * blockDim.x + threadIdx.x;
  if (i >= N * CDIM) return;
  int c = i & (CDIM - 1);
  float v = x[i] + bA[c] + bB[c];
  x[i] = v > 0.f ? v : 0.f;
}

// ---------------------------------------------------------------------------
// Host-side orchestration
// ---------------------------------------------------------------------------
static void run_etype(const float* fsrc, int Ns, const float* fdst, int Nd,
                      const int* src, const int* dst, int E,
                      const float* W, const float* al, const float* ar,
                      float* hs, float* el, float* er, unsigned* emax,
                      float* denom, float* war, float* accum, hipStream_t stream) {
  // projection of source features (the only big GEMM): hs = fsrc @ W
  dim3 ggrid(Ns / 16, CDIM / 128);               // Ns is a multiple of 16
  k_gemm_wmma_f32<<<ggrid, 256, 0, stream>>>(fsrc, W, hs, Ns);
  k_war<<<(CDIM * HH + 255) / 256, 256, 0, stream>>>(W, ar, war);
  k_el<<<(Ns * HH + 255) / 256, 256, 0, stream>>>(hs, al, el, Ns);
  k_er<<<(Nd * HH + 255) / 256, 256, 0, stream>>>(fdst, war, er, Nd);
  k_init_emax<<<(Nd * HH + 255) / 256, 256, 0, stream>>>(emax, Nd * HH);
  k_zero<<<(Nd * HH + 255) / 256, 256, 0, stream>>>(denom, Nd * HH);
  k_edge_max<<<(E + 255) / 256, 256, 0, stream>>>(src, dst, el, er, emax, E);
  k_edge_denom<<<(E + 255) / 256, 256, 0, stream>>>(src, dst, el, er, emax, denom, E);
  k_edge_scatter<<<E, 256, 0, stream>>>(src, dst, el, er, emax, denom, hs, accum, E);
}

extern "C" void kernel_launch(void* const* d_in, const int* in_sizes, int n_in,
                              void* d_out, int out_size, void* d_ws, size_t ws_size,
                              hipStream_t stream) {
  const float* feat_doc = (const float*)d_in[0];
  const float* feat_lab = (const float*)d_in[1];
  const float* W1  = (const float*)d_in[2];
  const float* al1 = (const float*)d_in[3];
  const float* ar1 = (const float*)d_in[4];
  const float* b1  = (const float*)d_in[5];
  const float* W2  = (const float*)d_in[6];
  const float* al2 = (const float*)d_in[7];
  const float* ar2 = (const float*)d_in[8];
  const float* b2  = (const float*)d_in[9];
  const int* g0[8];
  const int* g1[8];
  for (int i = 0; i < 8; ++i) g0[i] = (const int*)d_in[10 + i];
  for (int i = 0; i < 8; ++i) g1[i] = (const int*)d_in[18 + i];

  const int Ndoc = in_sizes[0] / CDIM;
  const int Nlab = in_sizes[1] / CDIM;
  const int E0_dl = in_sizes[10], E0_ld = in_sizes[12], E0_ll = in_sizes[14], E0_dd = in_sizes[16];
  const int E1_dl = in_sizes[18], E1_ld = in_sizes[20], E1_ll = in_sizes[22], E1_dd = in_sizes[24];

  // ---- workspace layout (floats) ----
  float* w = (float*)d_ws;
  float* hs      = w; w += (size_t)Ndoc * CDIM;     // src projections (reused)
  float* acc_doc = w; w += (size_t)Ndoc * CDIM;     // layer-1 doc accumulator / layer-2 input
  float* acc_lab = w; w += (size_t)Nlab * CDIM;     // layer-1 label accumulator / layer-2 input
  float* el      = w; w += (size_t)Ndoc * HH;
  float* er      = w; w += (size_t)Ndoc * HH;
  unsigned* emax = (unsigned*)w; w += (size_t)Ndoc * HH;
  float* denom   = w; w += (size_t)Ndoc * HH;
  float* war     = w; w += CDIM * HH;

  float* out_doc = (float*)d_out;
  float* out_lab = out_doc + (size_t)Ndoc * CDIM;

  const size_t WS = (size_t)CDIM * CDIM;    // per-etype W stride
  const size_t AS = (size_t)HH * DDIM;      // per-etype al/ar stride (== CDIM)
  const size_t BS = (size_t)CDIM;           // per-etype bias stride

  // ================= Layer 1 (graph g0, inputs = raw features) =============
  k_zero<<<((Ndoc * CDIM) + 255) / 256, 256, 0, stream>>>(acc_doc, Ndoc * CDIM);
  k_zero<<<((Nlab * CDIM) + 255) / 256, 256, 0, stream>>>(acc_lab, Nlab * CDIM);
  // e=0 doc->label
  run_etype(feat_doc, Ndoc, feat_lab, Nlab, g0[0], g0[1], E0_dl,
            W1 + 0 * WS, al1 + 0 * AS, ar1 + 0 * AS,
            hs, el, er, emax, denom, war, acc_lab, stream);
  // e=2 label->label
  run_etype(feat_lab, Nlab, feat_lab, Nlab, g0[4], g0[5], E0_ll,
            W1 + 2 * WS, al1 + 2 * AS, ar1 + 2 * AS,
            hs, el, er, emax, denom, war, acc_lab, stream);
  // e=1 label->doc
  run_etype(feat_lab, Nlab, feat_doc, Ndoc, g0[2], g0[3], E0_ld,
            W1 + 1 * WS, al1 + 1 * AS, ar1 + 1 * AS,
            hs, el, er, emax, denom, war, acc_doc, stream);
  // e=3 doc->doc
  run_etype(feat_doc, Ndoc, feat_doc, Ndoc, g0[6], g0[7], E0_dd,
            W1 + 3 * WS, al1 + 3 * AS, ar1 + 3 * AS,
            hs, el, er, emax, denom, war, acc_doc, stream);
  // bias(+both contributing etypes) + ReLU, in place -> layer-2 inputs
  k_finalize<<<((Ndoc * CDIM) + 255) / 256, 256, 0, stream>>>(acc_doc, b1 + 1 * BS, b1 + 3 * BS, Ndoc);
  k_finalize<<<((Nlab * CDIM) + 255) / 256, 256, 0, stream>>>(acc_lab, b1 + 0 * BS, b1 + 2 * BS, Nlab);

  // ================= Layer 2 (graph g1, accumulate straight into d_out) ====
  k_zero<<<((Ndoc * CDIM) + 255) / 256, 256, 0, stream>>>(out_doc, Ndoc * CDIM);
  k_zero<<<((Nlab * CDIM) + 255) / 256, 256, 0, stream>>>(out_lab, Nlab * CDIM);
  run_etype(acc_doc, Ndoc, acc_lab, Nlab, g1[0], g1[1], E1_dl,
            W2 + 0 * WS, al2 + 0 * AS, ar2 + 0 * AS,
            hs, el, er, emax, denom, war, out_lab, stream);
  run_etype(acc_lab, Nlab, acc_lab, Nlab, g1[4], g1[5], E1_ll,
            W2 + 2 * WS, al2 + 2 * AS, ar2 + 2 * AS,
            hs, el, er, emax, denom, war, out_lab, stream);
  run_etype(acc_lab, Nlab, acc_doc, Ndoc, g1[2], g1[3], E1_ld,
            W2 + 1 * WS, al2 + 1 * AS, ar2 + 1 * AS,
            hs, el, er, emax, denom, war, out_doc, stream);
  run_etype(acc_doc, Ndoc, acc_doc, Ndoc, g1[6], g1[7], E1_dd,
            W2 + 3 * WS, al2 + 3 * AS, ar2 + 3 * AS,
            hs, el, er, emax, denom, war, out_doc, stream);
  k_finalize<<<((Ndoc * CDIM) + 255) / 256, 256, 0, stream>>>(out_doc, b2 + 1 * BS, b2 + 3 * BS, Ndoc);
  k_finalize<<<((Nlab * CDIM) + 255) / 256, 256, 0, stream>>>(out_lab, b2 + 0 * BS, b2 + 2 * BS, Nlab);
}